// F1_Loss_83081847374037
// MI455X (gfx1250) — compile-verified
//
#include <hip/hip_runtime.h>
#include <hip/hip_bf16.h>

#define NCLS 46
#define EPSF 1e-7f

typedef __attribute__((ext_vector_type(2))) float v2f;
typedef __attribute__((ext_vector_type(8))) float v8f;

// ws layout (floats): [0..47] colsum, [48..95] tp, [96..143] counts
__global__ void f1_init_ws(float* ws) {
    int i = blockIdx.x * blockDim.x + threadIdx.x;
    if (i < 144) ws[i] = 0.0f;
}

// Column sums via V_WMMA_F32_16X16X4_F32 with all-ones A.
// Class mapping (A all-ones => any K permutation sums correctly):
//   class 2n+0 (n=0..15)  -> group0, slot n      class 2n+1 -> group1, slot n
//   class 32+2n (n=0..6)  -> group2, slot n      class 33+2n -> group3, slot n
// Lane loads float2 {2n,2n+1} per row => coalsced b64, zero predication.
// Lanes n>=7 of groups 2/3 load clamped addresses; the garbage lands in
// D columns 7..15 which are never read (matmul columns are independent).
__global__ void __launch_bounds__(256)
f1_main_kernel(const float* __restrict__ yp,
               const int* __restrict__ yt,
               float* __restrict__ ws, int N) {
    __shared__ float s_col[48];
    __shared__ float s_tp[NCLS];
    __shared__ float s_cnt[NCLS];
    for (int i = threadIdx.x; i < 48; i += blockDim.x) s_col[i] = 0.0f;
    for (int i = threadIdx.x; i < NCLS; i += blockDim.x) { s_tp[i] = 0.0f; s_cnt[i] = 0.0f; }
    __syncthreads();

    const int lane = threadIdx.x & 31;           // wave32
    const int h    = lane >> 4;                  // half-wave -> B element row select
    const int n    = lane & 15;                  // slot index (D column)
    const int gwave  = blockIdx.x * (blockDim.x >> 5) + (threadIdx.x >> 5);
    const int nwaves = gridDim.x * (blockDim.x >> 5);
    const int batchStride = nwaves * 8;

    v2f a; a[0] = 1.0f; a[1] = 1.0f;             // all-ones 16x4 A matrix
    v8f acc0 = {}, acc1 = {}, acc2 = {}, acc3 = {};

    const int colA = 2 * n;
    const int colB = (32 + 2 * n < NCLS) ? (32 + 2 * n) : 32;  // clamped, 8B aligned

    auto doChunk = [&](int c2) {
        const int ra = (c2 << 2) + h;            // rows ra, ra+2 feed B elems 0,1
        const float* pA = yp + (size_t)ra * NCLS;
        const float* pB = pA + 2 * NCLS;
        const float2 xa = *(const float2*)(pA + colA);
        const float2 xb = *(const float2*)(pB + colA);
        const float2 ya = *(const float2*)(pA + colB);
        const float2 yb = *(const float2*)(pB + colB);
        __builtin_prefetch(pA + (size_t)batchStride * 4 * NCLS + lane * 23, 0, 0);
        v2f b0, b1, b2, b3;
        b0[0] = xa.x; b0[1] = xb.x;
        b1[0] = xa.y; b1[1] = xb.y;
        b2[0] = ya.x; b2[1] = yb.x;
        b3[0] = ya.y; b3[1] = yb.y;
        acc0 = __builtin_amdgcn_wmma_f32_16x16x4_f32(false, a, false, b0, (short)0, acc0, false, false);
        acc1 = __builtin_amdgcn_wmma_f32_16x16x4_f32(false, a, false, b1, (short)0, acc1, false, false);
        acc2 = __builtin_amdgcn_wmma_f32_16x16x4_f32(false, a, false, b2, (short)0, acc2, false, false);
        acc3 = __builtin_amdgcn_wmma_f32_16x16x4_f32(false, a, false, b3, (short)0, acc3, false, false);
    };

    const int nFull = N >> 2;                    // full 4-row chunks
    int ch = gwave;

    // main region: batches of 8 chunks, no predication in WMMA path
    for (; ch + 7 * nwaves < nFull; ch += batchStride) {
#pragma unroll
        for (int k = 0; k < 8; ++k) doChunk(ch + k * nwaves);
        // batched tp/count: lane L -> chunk ch+(L>>2)*nwaves, row (L&3); all lanes uniform
        const int r = ((ch + (lane >> 2) * nwaves) << 2) + (lane & 3);
        const int t = yt[r];
        atomicAdd(&s_tp[t], yp[(size_t)r * NCLS + t]);
        atomicAdd(&s_cnt[t], 1.0f);
    }
    // leftover full chunks (< 8 for this wave)
    for (; ch < nFull; ch += nwaves) {
        doChunk(ch);
        if (lane < 4) {
            const int r = (ch << 2) + lane;
            const int t = yt[r];
            atomicAdd(&s_tp[t], yp[(size_t)r * NCLS + t]);
            atomicAdd(&s_cnt[t], 1.0f);
        }
    }
    // tail rows (N % 4): one wave handles them scalar-style
    if (gwave == 0) {
        for (int r = nFull << 2; r < N; ++r) {
            for (int c = lane; c < NCLS; c += 32)
                atomicAdd(&s_col[c], yp[(size_t)r * NCLS + c]);
            if (lane == 0) {
                const int t = yt[r];
                atomicAdd(&s_tp[t], yp[(size_t)r * NCLS + t]);
                atomicAdd(&s_cnt[t], 1.0f);
            }
        }
    }

    // flush WMMA accumulators: D elem0 at lane n holds colsum of its class
    if (lane < 16) {
        atomicAdd(&s_col[2 * lane],     acc0[0]);
        atomicAdd(&s_col[2 * lane + 1], acc1[0]);
    }
    if (lane < 7) {
        atomicAdd(&s_col[32 + 2 * lane], acc2[0]);
        atomicAdd(&s_col[33 + 2 * lane], acc3[0]);
    }
    __syncthreads();
    for (int i = threadIdx.x; i < NCLS; i += blockDim.x) {
        atomicAdd(&ws[i],      s_col[i]);
        atomicAdd(&ws[48 + i], s_tp[i]);
        atomicAdd(&ws[96 + i], s_cnt[i]);
    }
}

__global__ void f1_finalize(const float* __restrict__ ws, float* __restrict__ out) {
    if (blockIdx.x == 0 && threadIdx.x == 0) {
        float s = 0.0f;
        for (int c = 0; c < NCLS; ++c) {
            const float colsum = ws[c];        // tp + fp
            const float tp     = ws[48 + c];
            const float cnt    = ws[96 + c];   // tp + fn
            const float prec = tp / (colsum + EPSF);
            const float rec  = tp / (cnt + EPSF);
            float f1 = 2.0f * prec * rec / (prec + rec + EPSF);
            f1 = fminf(fmaxf(f1, EPSF), 1.0f - EPSF);
            s += f1;
        }
        out[0] = 1.0f - s / (float)NCLS;
    }
}

extern "C" void kernel_launch(void* const* d_in, const int* in_sizes, int n_in,
                              void* d_out, int out_size, void* d_ws, size_t ws_size,
                              hipStream_t stream) {
    const float* yp = (const float*)d_in[0];
    const int*   yt = (const int*)d_in[1];
    const int N = in_sizes[1];                 // rows (y_true length)
    float* ws = (float*)d_ws;

    f1_init_ws<<<1, 256, 0, stream>>>(ws);
    f1_main_kernel<<<1024, 256, 0, stream>>>(yp, yt, ws, N);
    f1_finalize<<<1, 64, 0, stream>>>(ws, (float*)d_out);
}